// NetStackedHourglass_2_37177236914283
// MI455X (gfx1250) — compile-verified
//
#include <hip/hip_runtime.h>
#include <hip/hip_bf16.h>

// ---------------------------------------------------------------------------
// NetStackedHourglass keypoint head for MI455X (gfx1250).
// Memory-bound streaming reduction (~587 MB read -> ~25 us at 23.3 TB/s).
// CDNA5 async global->LDS loads (ASYNCcnt) with a ping-pong double buffer:
// issue tile t+1, s_wait_asynccnt 7 (retire tile t only), consume tile t.
// ---------------------------------------------------------------------------

#define RES   64
#define NPIX  4096          // RES*RES
#define NCH   20
#define TILE  1024          // pixels per tile
#define NTILE 4             // NPIX / TILE
#define TPB   256           // threads per block (8 wave32)
#define BUFSZ (7 * TILE)    // floats per tile buffer: m,fd,bd (1K each) + fv,bv (2K each)

// ---- CDNA5 async global->LDS copy (16 bytes per lane) ----------------------
typedef __attribute__((__vector_size__(4 * sizeof(int)))) int v4i;
typedef __attribute__((address_space(1))) v4i* global_v4i_ptr;
typedef __attribute__((address_space(3))) v4i* lds_v4i_ptr;

__device__ __forceinline__ void async_copy_b128(void* lds, const void* gptr) {
#if defined(__has_builtin) && __has_builtin(__builtin_amdgcn_global_load_async_to_lds_b128)
  __builtin_amdgcn_global_load_async_to_lds_b128(
      (global_v4i_ptr)(gptr),
      (lds_v4i_ptr)(lds),
      /*offset=*/0, /*cpol=*/0);
#else
  unsigned lds_off = (unsigned)(unsigned long long)lds;
  asm volatile("global_load_async_to_lds_b128 %0, %1, off"
               :: "v"(lds_off), "v"(gptr)
               : "memory");
#endif
}

template <int N>
__device__ __forceinline__ void wait_async_le() {
#if defined(__has_builtin) && __has_builtin(__builtin_amdgcn_s_wait_asynccnt)
  __builtin_amdgcn_s_wait_asynccnt(N);
#else
  asm volatile("s_wait_asynccnt %0" :: "i"(N) : "memory");
#endif
}

// Issue the 7 async b128 loads (per thread) for one 1024-pixel tile.
__device__ __forceinline__ void issue_tile(
    float* buf, int tid, int t,
    const float* m_base, const float* fd_base, const float* bd_base,
    const float* fv_base, const float* bv_base)
{
  const int soff = t * TILE + 4 * tid;       // scalar-stream float offset
  const int voff = t * 2 * TILE + 8 * tid;   // vector-stream float offset
  async_copy_b128(buf + 0 * TILE + 4 * tid,     m_base  + soff);
  async_copy_b128(buf + 1 * TILE + 4 * tid,     fd_base + soff);
  async_copy_b128(buf + 2 * TILE + 4 * tid,     bd_base + soff);
  async_copy_b128(buf + 3 * TILE + 8 * tid,     fv_base + voff);
  async_copy_b128(buf + 3 * TILE + 8 * tid + 4, fv_base + voff + 4);
  async_copy_b128(buf + 5 * TILE + 8 * tid,     bv_base + voff);
  async_copy_b128(buf + 5 * TILE + 8 * tid + 4, bv_base + voff + 4);
}

// ---------------------------------------------------------------------------
// Kernel 1: per (batch, channel) masked reduction over 4096 pixels.
// Writes 5 floats per channel to ws: [msum, Fy, Fx, By, Bx] (F/B normalized).
// ---------------------------------------------------------------------------
__global__ __launch_bounds__(TPB) void hourglass_reduce_kernel(
    const float* __restrict__ front_vec,  // [B,40,64,64]
    const float* __restrict__ front_dis,  // [B,20,64,64]
    const float* __restrict__ back_vec,   // [B,40,64,64]
    const float* __restrict__ back_dis,   // [B,20,64,64]
    const float* __restrict__ ske_mask,   // [B,20,64,64]
    float* __restrict__ ws)               // [B*20*5]
{
  __shared__ float s_buf[2][BUFSZ];       // 2 x 28 KB ping-pong tile buffers
  __shared__ float s_red[8 * 5];

  const int bc  = blockIdx.x;             // 0 .. B*20-1
  const int b   = bc / NCH;
  const int c   = bc % NCH;
  const int tid = threadIdx.x;

  const float* m_base  = ske_mask  + ((size_t)b * NCH + c) * NPIX;
  const float* fd_base = front_dis + ((size_t)b * NCH + c) * NPIX;
  const float* bd_base = back_dis  + ((size_t)b * NCH + c) * NPIX;
  // vec channel c occupies 2*NPIX contiguous floats after row-major reshape
  const float* fv_base = front_vec + (size_t)b * (2 * NCH * NPIX) + (size_t)c * (2 * NPIX);
  const float* bv_base = back_vec  + (size_t)b * (2 * NCH * NPIX) + (size_t)c * (2 * NPIX);

  float a_m = 0.f, aFy = 0.f, aFx = 0.f, aBy = 0.f, aBx = 0.f;

  // Prologue: start tile 0 into buffer 0.
  issue_tile(s_buf[0], tid, 0, m_base, fd_base, bd_base, fv_base, bv_base);

#pragma unroll
  for (int t = 0; t < NTILE; ++t) {
    if (t + 1 < NTILE) {
      // Prefetch next tile into the other buffer (7 more async ops in flight).
      issue_tile(s_buf[(t + 1) & 1], tid, t + 1,
                 m_base, fd_base, bd_base, fv_base, bv_base);
      wait_async_le<7>();   // retire tile t's 7 loads; tile t+1 stays in flight
    } else {
      wait_async_le<0>();   // last tile: drain everything
    }
    __syncthreads();        // all waves' tile-t LDS writes visible

    const float* buf = s_buf[t & 1];
    const float4 m4  = ((const float4*)(buf + 0 * TILE))[tid];
    const float4 fd4 = ((const float4*)(buf + 1 * TILE))[tid];
    const float4 bd4 = ((const float4*)(buf + 2 * TILE))[tid];
    const float4 fv0 = ((const float4*)(buf + 3 * TILE))[2 * tid];
    const float4 fv1 = ((const float4*)(buf + 3 * TILE))[2 * tid + 1];
    const float4 bv0 = ((const float4*)(buf + 5 * TILE))[2 * tid];
    const float4 bv1 = ((const float4*)(buf + 5 * TILE))[2 * tid + 1];

    const float mm [4] = { m4.x,  m4.y,  m4.z,  m4.w  };
    const float fdd[4] = { fd4.x, fd4.y, fd4.z, fd4.w };
    const float bdd[4] = { bd4.x, bd4.y, bd4.z, bd4.w };
    const float fvy[4] = { fv0.x, fv0.z, fv1.x, fv1.z };
    const float fvx[4] = { fv0.y, fv0.w, fv1.y, fv1.w };
    const float bvy[4] = { bv0.x, bv0.z, bv1.x, bv1.z };
    const float bvx[4] = { bv0.y, bv0.w, bv1.y, bv1.w };

    const int pix0 = t * TILE + 4 * tid;
#pragma unroll
    for (int j = 0; j < 4; ++j) {
      const int pix = pix0 + j;
      const float y = (float)(pix >> 6);
      const float x = (float)(pix & (RES - 1));
      const float w = mm[j];
      a_m += w;
      aFy += w * fmaf(fvy[j] * fdd[j], (float)RES, y);
      aFx += w * fmaf(fvx[j] * fdd[j], (float)RES, x);
      aBy += w * fmaf(bvy[j] * bdd[j], (float)RES, y);
      aBx += w * fmaf(bvx[j] * bdd[j], (float)RES, x);
    }
    __syncthreads();        // all waves done reading buf before it is reused
  }

  // wave32 tree reduction
#pragma unroll
  for (int o = 16; o > 0; o >>= 1) {
    a_m += __shfl_xor(a_m, o, 32);
    aFy += __shfl_xor(aFy, o, 32);
    aFx += __shfl_xor(aFx, o, 32);
    aBy += __shfl_xor(aBy, o, 32);
    aBx += __shfl_xor(aBx, o, 32);
  }
  const int wave = tid >> 5;
  if ((tid & 31) == 0) {
    s_red[wave * 5 + 0] = a_m;
    s_red[wave * 5 + 1] = aFy;
    s_red[wave * 5 + 2] = aFx;
    s_red[wave * 5 + 3] = aBy;
    s_red[wave * 5 + 4] = aBx;
  }
  __syncthreads();
  if (tid == 0) {
    float sm = 0.f, fy = 0.f, fx = 0.f, by = 0.f, bx = 0.f;
#pragma unroll
    for (int w = 0; w < 8; ++w) {
      sm += s_red[w * 5 + 0];
      fy += s_red[w * 5 + 1];
      fx += s_red[w * 5 + 2];
      by += s_red[w * 5 + 3];
      bx += s_red[w * 5 + 4];
    }
    const float inv = 1.0f / (sm + 1e-6f);
    float* o = ws + (size_t)bc * 5;
    o[0] = sm;
    o[1] = fy * inv;
    o[2] = fx * inv;
    o[3] = by * inv;
    o[4] = bx * inv;
  }
}

// ---------------------------------------------------------------------------
// Kernel 2: per-batch keypoint assembly, out[b, 21, 2] = kp * 4
// ws channel layout: [msum, Fy, Fx, By, Bx]
// ---------------------------------------------------------------------------
__global__ __launch_bounds__(TPB) void hourglass_kp_kernel(
    const float* __restrict__ ws, float* __restrict__ out, int B)
{
  const int b = blockIdx.x * blockDim.x + threadIdx.x;
  if (b >= B) return;
  const float* w = ws + (size_t)b * (NCH * 5);

  float kp[21][2];
  float r0 = 0.f, r1 = 0.f;
#pragma unroll
  for (int i = 0; i < 5; ++i) {
    const float* wc = w + (4 * i) * 5;
    if (wc[0] != 0.0f) { r0 += wc[3]; r1 += wc[4]; }
  }
  kp[0][0] = r0 * 0.2f;
  kp[0][1] = r1 * 0.2f;

#pragma unroll
  for (int i = 0; i < 5; ++i) {
    const float* c0 = w + (4 * i + 0) * 5;
    const float* c1 = w + (4 * i + 1) * 5;
    const float* c2 = w + (4 * i + 2) * 5;
    const float* c3 = w + (4 * i + 3) * 5;
    kp[4 * i + 1][0] = c3[1];
    kp[4 * i + 1][1] = c3[2];
    kp[4 * i + 2][0] = (c2[1] + c3[3]) * 0.5f;
    kp[4 * i + 2][1] = (c2[2] + c3[4]) * 0.5f;
    kp[4 * i + 3][0] = (c1[1] + c2[3]) * 0.5f;
    kp[4 * i + 3][1] = (c1[2] + c2[4]) * 0.5f;
    kp[4 * i + 4][0] = (c0[1] + c1[3]) * 0.5f;
    kp[4 * i + 4][1] = (c0[2] + c1[4]) * 0.5f;
  }

  float* o = out + (size_t)b * 42;
#pragma unroll
  for (int j = 0; j < 21; ++j) {
    o[2 * j + 0] = kp[j][0] * 4.0f;
    o[2 * j + 1] = kp[j][1] * 4.0f;
  }
}

// ---------------------------------------------------------------------------
extern "C" void kernel_launch(void* const* d_in, const int* in_sizes, int n_in,
                              void* d_out, int out_size, void* d_ws, size_t ws_size,
                              hipStream_t stream) {
  const float* front_vec = (const float*)d_in[0];
  const float* front_dis = (const float*)d_in[1];
  const float* back_vec  = (const float*)d_in[2];
  const float* back_dis  = (const float*)d_in[3];
  const float* ske_mask  = (const float*)d_in[4];

  const int B = in_sizes[1] / (NCH * NPIX);   // front_dis is [B,20,64,64]
  float* ws = (float*)d_ws;                   // B*20*5 floats of scratch

  hourglass_reduce_kernel<<<B * NCH, TPB, 0, stream>>>(
      front_vec, front_dis, back_vec, back_dis, ske_mask, ws);

  hourglass_kp_kernel<<<(B + TPB - 1) / TPB, TPB, 0, stream>>>(
      ws, (float*)d_out, B);
}